// GaussianBlur_27333171872128
// MI455X (gfx1250) — compile-verified
//
#include <hip/hip_runtime.h>

// Gaussian blur sigma=2, 13-tap separable, NHWC (32,384,384,16) f32, SAME pad.
//
// Roofline: >=604MB HBM traffic -> ~26us floor @23.3TB/s -> memory bound.
// Separable passes fused through LDS; MACs on the matrix pipe via exact-f32
// V_WMMA_F32_16X16X4_F32: D(16x16) = band(16x28) x data(28x16), K=28 -> 7
// accumulating WMMAs per tile row/column. C=16 (contiguous NHWC innermost dim)
// maps to WMMA N -> all global traffic is coalesced 64B segments per half-wave.
//
// v2: padding via clamped-address load + 0/1 mask multiply (no predicated
// loads / EXEC gymnastics), uniform interior fast path without masks,
// unroll-2 row loop for load/WMMA overlap, nontemporal output stores.

typedef __attribute__((ext_vector_type(2))) float v2f;
typedef __attribute__((ext_vector_type(8))) float v8f;

#define N_IMG 32
#define H_IMG 384
#define W_IMG 384
#define C_IMG 16
#define TS    16          // tile size in x, y; equals channel count (N dim)
#define HALO  6           // (13-1)/2
#define EXT   28          // TS + 2*HALO
#define NG    7           // ceil(EXT / 4) WMMA K-groups
#define LSTR  19          // LDS inner stride (odd; 8*19%64=24, 32*19%64=32 -> conflict-free)

__global__ __launch_bounds__(32)
void gauss13_wmma(const float* __restrict__ xin, float* __restrict__ out) {
  const int x0   = blockIdx.x * TS;
  const int y0   = blockIdx.y * TS;
  const int n    = blockIdx.z;
  const int lane = threadIdx.x;     // 0..31 (wave32)
  const int lnLo = lane & 15;       // -> N index (channel) / M index
  const int lnHi = lane >> 4;       // 0/1 half-wave

  __shared__ float Th[EXT * TS * LSTR];   // horizontal-pass result [j_y][x_local][c]

  // --- 1D gaussian weights (sigma=2): w[i] = exp(-(i-6)^2/8) / sum ---
  float s = 0.f;
  #pragma unroll
  for (int i = 0; i < 13; ++i) {
    float d = (float)(i - 6);
    s += __expf(-d * d * 0.125f);
  }
  const float inv_s = 1.f / s;

  // --- Constant A operand: 16x28 band matrix A[m,j] = w[j-m], 0<=j-m<=12 ---
  // A layout (f32 16x16x4): slot v: element (M = lane%16, K = 4g + v + 2*(lane/16))
  // Also precompute per-lane clamped input-column offsets + pad masks
  // (loop-invariant across all 28 rows of stage 1).
  v2f   a[NG];
  int   off[NG][2];
  float msk[NG][2];
  #pragma unroll
  for (int g = 0; g < NG; ++g) {
    #pragma unroll
    for (int v = 0; v < 2; ++v) {
      int K = 4 * g + v + 2 * lnHi;
      int d = K - lnLo;                        // band offset 0..12
      float o = (float)(d - 6);
      a[g][v] = (d >= 0 && d <= 12) ? __expf(-o * o * 0.125f) * inv_s : 0.f;
      int x_in = x0 - HALO + K;                // halo-local -> global x
      int xc = min(max(x_in, 0), W_IMG - 1);   // always-valid address
      off[g][v] = xc * C_IMG + lnLo;
      msk[g][v] = (x_in >= 0 && x_in < W_IMG) ? 1.f : 0.f;
    }
  }

  const size_t base_n = (size_t)n * H_IMG * W_IMG * C_IMG;
  const bool interior = (x0 >= HALO) && (x0 + TS + HALO <= W_IMG);  // uniform

  // ---- Stage 1: horizontal pass over rows y0-6 .. y0+21 ----
  #pragma unroll 2
  for (int jy = 0; jy < EXT; ++jy) {
    const int y_in = y0 - HALO + jy;
    v8f acc = {};
    if (y_in >= 0 && y_in < H_IMG) {          // uniform branch: EXEC stays full
      const float* row = xin + base_n + (size_t)y_in * (W_IMG * C_IMG);
      if (interior) {                          // uniform: ~92% of tiles, no masks
        #pragma unroll
        for (int g = 0; g < NG; ++g) {
          v2f b;
          b[0] = row[off[g][0]];
          b[1] = row[off[g][1]];
          acc = __builtin_amdgcn_wmma_f32_16x16x4_f32(false, a[g], false, b,
                                                      (short)0, acc, false, false);
        }
      } else {                                 // border: zero-pad via mask mul
        #pragma unroll
        for (int g = 0; g < NG; ++g) {
          v2f b;
          b[0] = row[off[g][0]] * msk[g][0];
          b[1] = row[off[g][1]] * msk[g][1];
          acc = __builtin_amdgcn_wmma_f32_16x16x4_f32(false, a[g], false, b,
                                                      (short)0, acc, false, false);
        }
      }
    }
    // D layout: VGPR p -> (M = p + 8*lnHi, N = lnLo); write row jy to LDS
    #pragma unroll
    for (int p = 0; p < 8; ++p) {
      int xl = p + 8 * lnHi;
      Th[(jy * TS + xl) * LSTR + lnLo] = acc[p];
    }
  }

  __syncthreads();   // single wave: cheap; orders LDS writes before reads

  // ---- Stage 2: vertical pass, one 16(y) x 16(c) WMMA chain per x column ----
  for (int xl = 0; xl < TS; ++xl) {
    v8f acc = {};
    #pragma unroll
    for (int g = 0; g < NG; ++g) {
      v2f b;                                   // B[j, c] = Th[j][xl][c]
      #pragma unroll
      for (int v = 0; v < 2; ++v) {
        int j = 4 * g + v + 2 * lnHi;
        b[v] = Th[(j * TS + xl) * LSTR + lnLo];
      }
      acc = __builtin_amdgcn_wmma_f32_16x16x4_f32(false, a[g], false, b,
                                                  (short)0, acc, false, false);
    }
    #pragma unroll
    for (int p = 0; p < 8; ++p) {
      int yl = p + 8 * lnHi;
      __builtin_nontemporal_store(
          acc[p],
          &out[base_n + ((size_t)(y0 + yl) * W_IMG + (x0 + xl)) * C_IMG + lnLo]);
    }
  }
}

extern "C" void kernel_launch(void* const* d_in, const int* in_sizes, int n_in,
                              void* d_out, int out_size, void* d_ws, size_t ws_size,
                              hipStream_t stream) {
  (void)in_sizes; (void)n_in; (void)out_size; (void)d_ws; (void)ws_size;
  const float* x = (const float*)d_in[0];
  float* out = (float*)d_out;
  dim3 grid(W_IMG / TS, H_IMG / TS, N_IMG);   // 24 x 24 x 32 = 18432 waves
  dim3 block(32);                              // one wave32 per tile
  gauss13_wmma<<<grid, block, 0, stream>>>(x, out);
}